// Advect_43954695307581
// MI455X (gfx1250) — compile-verified
//
#include <hip/hip_runtime.h>
#include <cstdint>

#define SLOPE_TH   2.0f
#define L_LINE     8192
#define OUT_L      (L_LINE - 4)      // 8188
#define TILE       1024
#define LDS_ELEMS  (TILE + 4)        // 1028 floats staged per array

typedef float v4f __attribute__((ext_vector_type(4)));

// Async 128-bit global->LDS copy, non-temporal (each input byte used once;
// 402MB stream vs 192MB L2 -> NT avoids cache churn). Tracked by ASYNCcnt.
__device__ __forceinline__ void async_copy_b128_nt(unsigned lds_byte_addr,
                                                   const float* gptr) {
    asm volatile("global_load_async_to_lds_b128 %0, %1, off th:TH_LOAD_NT"
                 :: "v"(lds_byte_addr), "v"(gptr) : "memory");
}

__device__ __forceinline__ float minmod3(float a, float b, float c) {
    // matches jnp: fmin<0 ? min(fmax,0) : fmin
    float mn = fminf(fminf(a, b), c);
    float mx = fmaxf(fmaxf(a, b), c);
    return (mn < 0.0f) ? fminf(mx, 0.0f) : mn;
}

__global__ __launch_bounds__(256) void advect_minmod_kernel(
    const float* __restrict__ rho,
    const float* __restrict__ v,
    float* __restrict__ out)
{
    __shared__ float srho[LDS_ELEMS];
    __shared__ float sv[LDS_ELEMS];

    const int tid  = threadIdx.x;
    const int tile = blockIdx.x;          // 0..7
    const int line = blockIdx.y;          // 0..4095  (b*M + m)
    const int base = tile * TILE;         // line-local element/output base
    const long long gline = (long long)line * L_LINE;
    const float* rline = rho + gline;
    const float* vline = v   + gline;

    // LDS byte offsets: low 32 bits of the flat shared address are the LDS
    // address (ISA aperture rule).
    const unsigned lds_rho0 = (unsigned)(unsigned long long)(void*)srho;
    const unsigned lds_v0   = (unsigned)(unsigned long long)(void*)sv;

    // Stage [base, base+1027] of rho and v into LDS: one 16B chunk per thread,
    // thread 0 also copies the 4-element halo chunk. Clamp the global element
    // index so the last tile never reads past the buffer; clamped (duplicated)
    // LDS slots lie beyond any consumed rel index.
    {
        int e = base + 4 * tid;
        if (e > L_LINE - 4) e = L_LINE - 4;
        async_copy_b128_nt(lds_rho0 + 16u * (unsigned)tid, rline + e);
        async_copy_b128_nt(lds_v0   + 16u * (unsigned)tid, vline + e);
        if (tid == 0) {
            int eh = base + 4 * 256;                  // halo chunk (rel 1024..1027)
            if (eh > L_LINE - 4) eh = L_LINE - 4;
            async_copy_b128_nt(lds_rho0 + 16u * 256u, rline + eh);
            async_copy_b128_nt(lds_v0   + 16u * 256u, vline + eh);
        }
    }
    asm volatile("s_wait_asynccnt 0x0" ::: "memory");
    __syncthreads();

    const int n_out = min(TILE, OUT_L - base);  // 1024, or 1020 for last tile
    const int k0 = tid * 4;                     // 4 contiguous outputs/thread
    if (k0 < n_out) {
        // rel indices 0..7 of flux/v needed for outputs k0..k0+3
        v4f r0 = *reinterpret_cast<const v4f*>(&srho[k0]);
        v4f r1 = *reinterpret_cast<const v4f*>(&srho[k0 + 4]);
        v4f w0 = *reinterpret_cast<const v4f*>(&sv[k0]);
        v4f w1 = *reinterpret_cast<const v4f*>(&sv[k0 + 4]);

        float f[8]  = {r0.x*w0.x, r0.y*w0.y, r0.z*w0.z, r0.w*w0.w,
                       r1.x*w1.x, r1.y*w1.y, r1.z*w1.z, r1.w*w1.w};
        float vv[8] = {w0.x, w0.y, w0.z, w0.w, w1.x, w1.y, w1.z, w1.w};

        // half_slope rel i (i = 0..5): uses flux rel i..i+2
        float hs[6];
        #pragma unroll
        for (int i = 0; i < 6; ++i) {
            float c0 = SLOPE_TH * (f[i+1] - f[i]);
            float c1 = 0.5f     * (f[i+2] - f[i]);
            float c2 = SLOPE_TH * (f[i+2] - f[i+1]);
            hs[i] = 0.5f * minmod3(c0, c1, c2);
        }

        // flux_net rel j (j = 0..4), line-local index jg = base + k0 + j
        float fn[5];
        #pragma unroll
        for (int j = 0; j < 5; ++j) {
            const int jg = base + k0 + j;
            float fm = f[j+2] - hs[j+1];
            fm = (vv[j+2] >= 0.0f) ? 0.0f : fm;
            if (jg == OUT_L) fm = 0.0f;          // flux_minus[..., -1] = 0
            float fp = f[j+1] + hs[j];
            fp = (vv[j+1] <= 0.0f) ? 0.0f : fp;
            if (jg == 0) fp = 0.0f;              // flux_plus[..., 0] = 0
            fn[j] = fm + fp;
        }

        // out[k] = -(fn[k+1] - fn[k]);  write-once stream -> non-temporal store
        v4f o = { fn[0]-fn[1], fn[1]-fn[2], fn[2]-fn[3], fn[3]-fn[4] };
        float* op = out + (long long)line * OUT_L + base + k0;   // 16B aligned
        __builtin_nontemporal_store(o, reinterpret_cast<v4f*>(op));
    }
}

extern "C" void kernel_launch(void* const* d_in, const int* in_sizes, int n_in,
                              void* d_out, int out_size, void* d_ws, size_t ws_size,
                              hipStream_t stream) {
    (void)n_in; (void)d_ws; (void)ws_size; (void)out_size;
    const float* rho = (const float*)d_in[0];
    const float* v   = (const float*)d_in[1];
    // d_in[2] = axis (always 2 / last), d_in[3] = retain_padding (0): ignored.
    float* out = (float*)d_out;

    const int lines = in_sizes[0] / L_LINE;               // B*M = 4096
    dim3 grid((OUT_L + TILE - 1) / TILE, lines);          // (8, 4096)
    dim3 block(256);
    advect_minmod_kernel<<<grid, block, 0, stream>>>(rho, v, out);
}